// FocusClassificationLoss_65403761983603
// MI455X (gfx1250) — compile-verified
//
#include <hip/hip_runtime.h>
#include <hip/hip_bf16.h>
#include <math.h>

#define NROWS 8192
#define NC    80
#define KSTEPS 20          // 80 / 4 WMMA steps
#define KQUADS 10          // 80 / 8 b128 loads per fragment
#define GAMMA 2.0f
#define ALPHA 0.25f
#define MARGIN 0.5f
#define PAIR_EPS 1e-6f
#define NPART 1024         // partial blocks for K1 / K3

typedef __attribute__((ext_vector_type(2))) float f32x2;
typedef __attribute__((ext_vector_type(4))) float f32x4;
typedef __attribute__((ext_vector_type(8))) float f32x8;

// ---------------------------------------------------------------------------
// Kernel 1: focal-loss partial sums (block-staged, deterministic) + row norms
// 8 rows per block, one wave32 per row.
// ---------------------------------------------------------------------------
__global__ __launch_bounds__(256) void focal_sq_kernel(
    const float* __restrict__ pred, const int* __restrict__ target,
    float* __restrict__ sq, float* __restrict__ focalPart)
{
    const int lane = threadIdx.x & 31;
    const int wave = threadIdx.x >> 5;          // 0..7
    const int row  = blockIdx.x * 8 + wave;

    const float* pr = pred + (long long)row * NC;
    const int    tg = target[row];              // in [0, NC]; NC = background

    float facc = 0.0f;
    float ssq  = 0.0f;
    for (int c = lane; c < NC; c += 32) {
        const float x = pr[c];
        ssq += x * x;
        const float o  = (tg == c) ? 1.0f : 0.0f;
        const float p  = 1.0f / (1.0f + __expf(-x));
        const float pt = (1.0f - p) * o + p * (1.0f - o);
        const float fw = (ALPHA * o + (1.0f - ALPHA) * (1.0f - o)) * pt * pt;
        const float bce = fmaxf(x, 0.0f) - x * o + log1pf(__expf(-fabsf(x)));
        facc += bce * fw;
    }
    // wave32 reductions
    for (int off = 16; off > 0; off >>= 1) {
        facc += __shfl_down(facc, off, 32);
        ssq  += __shfl_down(ssq,  off, 32);
    }

    __shared__ float s_f[8];
    if (lane == 0) {
        sq[row]  = ssq;
        s_f[wave] = facc;
    }
    __syncthreads();
    if (threadIdx.x == 0) {
        float s = 0.0f;
        #pragma unroll
        for (int w = 0; w < 8; ++w) s += s_f[w];
        focalPart[blockIdx.x] = s;
    }
}

// ---------------------------------------------------------------------------
// Kernel 2: fused Gram-matrix (WMMA f32 16x16x4) + hardest pos/neg mining.
// Block = 256 threads (8 waves); block owns 16 anchor rows; each wave computes
// a 16x16 dot tile per iteration (8 waves -> 128 columns / iteration).
//
// K-permutation: for step pair (2t, 2t+1), lane-half hi covers logical
// k = 8t + 4*hi + {0..3}  ->  one 16B-aligned b128 load feeds two WMMA steps.
// (A and B use the same permutation; dot products are order-invariant.)
//
// Mining proxy: score = sq_j - 2*dot (monotone in distance per row); updates
// are branchless (v_cmp + v_cndmask), strict compares preserve
// first-occurrence argmin/argmax semantics.
// ---------------------------------------------------------------------------
__global__ __launch_bounds__(256) void mine_kernel(
    const float* __restrict__ pred, const int* __restrict__ target,
    const float* __restrict__ sq,
    int* __restrict__ posIdx, int* __restrict__ negIdx)
{
    const int lane = threadIdx.x & 31;
    const int wave = threadIdx.x >> 5;   // 0..7
    const int hi   = lane >> 4;          // 0/1 : K-slot / M-half selector
    const int n    = lane & 15;          // column within tile / anchor row for A
    const int ib   = blockIdx.x * 16;    // anchor row base

    // A fragments (persist across the j-loop): lane holds row (ib+n),
    // quad t = floats [8t+4*hi .. 8t+4*hi+3]; unpack to pairs per WMMA step.
    f32x2 a2[KSTEPS];
    {
        const float* arow = pred + (long long)(ib + n) * NC + 4 * hi;
        #pragma unroll
        for (int t = 0; t < KQUADS; ++t) {
            const f32x4 q = *reinterpret_cast<const f32x4*>(arow + 8 * t);
            a2[2 * t].x     = q.x;  a2[2 * t].y     = q.y;
            a2[2 * t + 1].x = q.z;  a2[2 * t + 1].y = q.w;
        }
    }

    // Per-lane anchor-row targets for the 8 accumulator slots (m = r + 8*hi)
    int ti[8];
    #pragma unroll
    for (int r = 0; r < 8; ++r) ti[r] = target[ib + r + 8 * hi];

    float minv[8], maxv[8];
    int   mini[8], maxi[8];
    #pragma unroll
    for (int r = 0; r < 8; ++r) {
        minv[r] =  INFINITY; mini[r] = 0;
        maxv[r] = -INFINITY; maxi[r] = 0;
    }

    for (int jb = 0; jb < NROWS; jb += 128) {
        const int   jcol = jb + wave * 16 + n;           // this lane's column
        const float sqj  = sq[jcol];
        const int   tj   = target[jcol];
        const float* brow = pred + (long long)jcol * NC + 4 * hi;

        // Phase 1: batch all B-fragment loads (distinct destinations -> one
        // clause, incremental s_wait_loadcnt draining, full MLP).
        f32x4 b4[KQUADS];
        #pragma unroll
        for (int t = 0; t < KQUADS; ++t)
            b4[t] = *reinterpret_cast<const f32x4*>(brow + 8 * t);

        // Phase 2: chained WMMA accumulation (two steps per quad).
        f32x8 acc = {};
        #pragma unroll
        for (int t = 0; t < KQUADS; ++t) {
            f32x2 blo; blo.x = b4[t].x; blo.y = b4[t].y;
            acc = __builtin_amdgcn_wmma_f32_16x16x4_f32(
                false, a2[2 * t], false, blo, (short)0, acc, false, false);
            f32x2 bhi; bhi.x = b4[t].z; bhi.y = b4[t].w;
            acc = __builtin_amdgcn_wmma_f32_16x16x4_f32(
                false, a2[2 * t + 1], false, bhi, (short)0, acc, false, false);
        }

        // acc[r] = dot(pred[ib + r + 8*hi], pred[jcol])
        #pragma unroll
        for (int r = 0; r < 8; ++r) {
            const int   i     = ib + r + 8 * hi;
            const float score = sqj - 2.0f * acc[r];
            const bool  same  = (ti[r] == tj);
            const bool  offd  = (i != jcol);

            // hardest (closest) positive: same class, j != i  (branchless)
            const float candP = (same & offd) ? score : INFINITY;
            const bool  updP  = candP < minv[r];
            mini[r] = updP ? jcol  : mini[r];
            minv[r] = updP ? candP : minv[r];

            // hardest (farthest) negative: different class  (branchless)
            const float candN = same ? -INFINITY : score;
            const bool  updN  = candN > maxv[r];
            maxi[r] = updN ? jcol  : maxi[r];
            maxv[r] = updN ? candN : maxv[r];
        }
    }

    // Stage per-lane candidates and merge with index tie-break (matches
    // jnp.argmin/argmax first-occurrence; all-masked rows resolve to 0).
    __shared__ float s_minv[8][16][16];
    __shared__ int   s_mini[8][16][16];
    __shared__ float s_maxv[8][16][16];
    __shared__ int   s_maxi[8][16][16];

    #pragma unroll
    for (int r = 0; r < 8; ++r) {
        const int m = r + 8 * hi;
        s_minv[wave][m][n] = minv[r]; s_mini[wave][m][n] = mini[r];
        s_maxv[wave][m][n] = maxv[r]; s_maxi[wave][m][n] = maxi[r];
    }
    __syncthreads();

    if (threadIdx.x < 16) {
        const int m = threadIdx.x;
        float bv =  INFINITY; int bi = 0;
        float wv = -INFINITY; int wi = 0;
        for (int w = 0; w < 8; ++w) {
            for (int q = 0; q < 16; ++q) {
                const float v1 = s_minv[w][m][q]; const int i1 = s_mini[w][m][q];
                const bool u1 = (v1 < bv) | ((v1 == bv) & (i1 < bi));
                bi = u1 ? i1 : bi;  bv = u1 ? v1 : bv;
                const float v2 = s_maxv[w][m][q]; const int i2 = s_maxi[w][m][q];
                const bool u2 = (v2 > wv) | ((v2 == wv) & (i2 < wi));
                wi = u2 ? i2 : wi;  wv = u2 ? v2 : wv;
            }
        }
        posIdx[ib + m] = bi;
        negIdx[ib + m] = wi;
    }
}

// ---------------------------------------------------------------------------
// Kernel 3: per-anchor triplet loss partials (8 rows/block, wave per row)
// ---------------------------------------------------------------------------
__global__ __launch_bounds__(256) void triplet_kernel(
    const float* __restrict__ pred,
    const int* __restrict__ posIdx, const int* __restrict__ negIdx,
    float* __restrict__ tripPart, float* __restrict__ cntPart)
{
    const int lane = threadIdx.x & 31;
    const int wave = threadIdx.x >> 5;
    const int row  = blockIdx.x * 8 + wave;

    const int pi = posIdx[row];
    const int ni = negIdx[row];
    const float* pr = pred + (long long)row * NC;
    const float* pp = pred + (long long)pi  * NC;
    const float* pn = pred + (long long)ni  * NC;

    float sap = 0.0f, san = 0.0f;
    for (int c = lane; c < NC; c += 32) {
        const float a = pr[c];
        const float d1 = a - pp[c] + PAIR_EPS;  sap += d1 * d1;
        const float d2 = a - pn[c] + PAIR_EPS;  san += d2 * d2;
    }
    for (int off = 16; off > 0; off >>= 1) {
        sap += __shfl_down(sap, off, 32);
        san += __shfl_down(san, off, 32);
    }

    __shared__ float s_sum[8];
    __shared__ float s_cnt[8];
    if (lane == 0) {
        const float d_ap = sqrtf(sap);
        const float d_an = sqrtf(san);
        const float per  = fmaxf(d_ap - d_an + MARGIN, 0.0f);
        const float vld  = (pi != ni) ? 1.0f : 0.0f;
        s_sum[wave] = per * vld;
        s_cnt[wave] = vld;
    }
    __syncthreads();
    if (threadIdx.x == 0) {
        float s = 0.0f, c = 0.0f;
        #pragma unroll
        for (int w = 0; w < 8; ++w) { s += s_sum[w]; c += s_cnt[w]; }
        tripPart[blockIdx.x] = s;
        cntPart[blockIdx.x]  = c;
    }
}

// ---------------------------------------------------------------------------
// Kernel 4: deterministic finalize (single block)
// ---------------------------------------------------------------------------
__global__ __launch_bounds__(256) void finalize_kernel(
    const float* __restrict__ focalPart, const float* __restrict__ tripPart,
    const float* __restrict__ cntPart, float* __restrict__ out)
{
    __shared__ float sf[256], st[256], sc[256];
    const int t = threadIdx.x;
    float f = 0.0f, s = 0.0f, c = 0.0f;
    for (int i = t; i < NPART; i += 256) {
        f += focalPart[i]; s += tripPart[i]; c += cntPart[i];
    }
    sf[t] = f; st[t] = s; sc[t] = c;
    __syncthreads();
    for (int off = 128; off > 0; off >>= 1) {
        if (t < off) { sf[t] += sf[t + off]; st[t] += st[t + off]; sc[t] += sc[t + off]; }
        __syncthreads();
    }
    if (t == 0) {
        out[0] = sf[0] / (float)(NROWS * NC);                       // FOCAL_W = 1
        const float cnt = sc[0];
        out[1] = (cnt > 0.0f) ? (st[0] / fmaxf(cnt, 1.0f)) : 0.0f;  // TRIPLET_W = 1
    }
}

// ---------------------------------------------------------------------------
extern "C" void kernel_launch(void* const* d_in, const int* in_sizes, int n_in,
                              void* d_out, int out_size, void* d_ws, size_t ws_size,
                              hipStream_t stream)
{
    const float* pred   = (const float*)d_in[0];
    const int*   target = (const int*)d_in[1];
    float*       out    = (float*)d_out;

    // ws layout (all 4-byte types, base is allocation-aligned)
    char* ws = (char*)d_ws;
    float* sq        = (float*)ws;                       ws += NROWS * sizeof(float);
    int*   posIdx    = (int*)ws;                         ws += NROWS * sizeof(int);
    int*   negIdx    = (int*)ws;                         ws += NROWS * sizeof(int);
    float* focalPart = (float*)ws;                       ws += NPART * sizeof(float);
    float* tripPart  = (float*)ws;                       ws += NPART * sizeof(float);
    float* cntPart   = (float*)ws;

    focal_sq_kernel<<<NROWS / 8, 256, 0, stream>>>(pred, target, sq, focalPart);
    mine_kernel<<<NROWS / 16, 256, 0, stream>>>(pred, target, sq, posIdx, negIdx);
    triplet_kernel<<<NROWS / 8, 256, 0, stream>>>(pred, posIdx, negIdx, tripPart, cntPart);
    finalize_kernel<<<1, 256, 0, stream>>>(focalPart, tripPart, cntPart, out);
}